// RenderingKaolin_54674933678446
// MI455X (gfx1250) — compile-verified
//
#include <hip/hip_runtime.h>
#include <math.h>

#define Vn 642
#define Fn 640
#define Hh 160
#define Ww 160
#define NFr 2
#define STEPS 8
#define HW (Hh*Ww)
#define NR (NFr*STEPS)          // 16 renders
#define FOCAL_F 2.4155736f      // 1/tan(1.57/4)
#define SIGMAINV_F 7000.0f
#define CAMZ 2.0f

typedef float v2f __attribute__((ext_vector_type(2)));
typedef float v8f __attribute__((ext_vector_type(8)));

__device__ __constant__ float G11[11] = {
    1.4867195e-06f, 1.3383022e-04f, 4.4318484e-03f, 5.3990967e-02f,
    2.4197073e-01f, 3.9894228e-01f, 2.4197073e-01f, 5.3990967e-02f,
    4.4318484e-03f, 1.3383022e-04f, 1.4867195e-06f};

// ---------------------------------------------------------------- setup ----
__device__ inline void quat_to_aa(const float* q, float* aa) {
    float w = q[0], x = q[1], y = q[2], z = q[3];
    float sn = sqrtf(x*x + y*y + z*z);
    float tt = 2.0f * ((w < 0.0f) ? atan2f(-sn, -w) : atan2f(sn, w));
    float k  = (sn > 1e-8f) ? (tt / fmaxf(sn, 1e-8f)) : 2.0f;
    aa[0] = x*k; aa[1] = y*k; aa[2] = z*k;
}

__device__ inline void rodrigues(const float* aa, float* R) {
    float th = sqrtf(aa[0]*aa[0] + aa[1]*aa[1] + aa[2]*aa[2]);
    float inv = 1.0f / fmaxf(th, 1e-8f);
    float x = aa[0]*inv, y = aa[1]*inv, z = aa[2]*inv;
    float s = sinf(th), c = cosf(th), cc = 1.0f - c;
    R[0] = 1.0f + cc*(-(y*y + z*z));
    R[1] = -s*z + cc*(x*y);
    R[2] =  s*y + cc*(x*z);
    R[3] =  s*z + cc*(x*y);
    R[4] = 1.0f + cc*(-(x*x + z*z));
    R[5] = -s*x + cc*(y*z);
    R[6] = -s*y + cc*(x*z);
    R[7] =  s*x + cc*(y*z);
    R[8] = 1.0f + cc*(-(x*x + y*y));
}

// Rbuf: per render r = f*8+t : 9 floats R, 3 floats translation (12 total)
__global__ void k_setup(const float* __restrict__ trans,
                        const float* __restrict__ quat,
                        float* __restrict__ Rbuf) {
    int f = threadIdx.x;
    if (f >= NFr) return;
    float aa[3], R0[9], Rs[9];
    quat_to_aa(quat + (f*2 + 1)*4, aa);
    rodrigues(aa, R0);
    quat_to_aa(quat + (f*2 + 0)*4, aa);
    aa[0] *= (1.0f/16.0f); aa[1] *= (1.0f/16.0f); aa[2] *= (1.0f/16.0f);
    rodrigues(aa, Rs);
    const float* tb = trans + (f*2 + 1)*3;
    const float* tv = trans + (f*2 + 0)*3;
    float R[9];
    #pragma unroll
    for (int i = 0; i < 9; ++i) R[i] = R0[i];
    for (int t = 0; t < STEPS; ++t) {
        float* o = Rbuf + (size_t)(f*STEPS + t)*12;
        #pragma unroll
        for (int i = 0; i < 9; ++i) o[i] = R[i];
        float ti = (float)t / 7.0f;
        o[9]  = tb[0] + ti*tv[0];
        o[10] = tb[1] + ti*tv[1];
        o[11] = tb[2] + ti*tv[2];
        float Rn[9];
        #pragma unroll
        for (int i = 0; i < 3; ++i)
            #pragma unroll
            for (int j = 0; j < 3; ++j)
                Rn[i*3+j] = R[i*3+0]*Rs[0*3+j] + R[i*3+1]*Rs[1*3+j] + R[i*3+2]*Rs[2*3+j];
        #pragma unroll
        for (int i = 0; i < 9; ++i) R[i] = Rn[i];
    }
}

// ------------------------------------------------------- face coefficients --
// bOp: per (render, group of 4 faces): 64 floats = the WMMA B operand,
// pre-swizzled to lane layout (VGPR0 = K{0,2}, VGPR1 = K{1,3} half-wave split):
//   float2 slot j    (j<16)  = (coef_x, coef_y) of column j
//   float2 slot 16+j         = (coef_0, 0)      of column j
// column j = (face_in_group)*4 + attr, attrs = {w0, w1, w2, z-plane}.
// Invalid faces (backfacing / degenerate) are folded in by setting the w0
// plane to (0, 0, -1e30): inside-test fails and sigmoid(7000*-1e30) == 0.
__global__ void k_facesetup(const float* __restrict__ Rbuf,
                            const float* __restrict__ uverts,
                            const int*   __restrict__ faces,
                            float* __restrict__ bOp) {
    int idx = blockIdx.x * blockDim.x + threadIdx.x;
    if (idx >= NR * Fn) return;
    int r = idx / Fn, face = idx % Fn;
    const float* Rt = Rbuf + (size_t)r*12;
    float fx[3], fy[3], fz3[3];
    #pragma unroll
    for (int k = 0; k < 3; ++k) {
        int vi = faces[face*3 + k];
        const float* u = uverts + (size_t)vi*3;
        fx[k]  = Rt[0]*u[0] + Rt[1]*u[1] + Rt[2]*u[2] + Rt[9];
        fy[k]  = Rt[3]*u[0] + Rt[4]*u[1] + Rt[5]*u[2] + Rt[10];
        fz3[k] = Rt[6]*u[0] + Rt[7]*u[1] + Rt[8]*u[2] + Rt[11] - CAMZ;
    }
    float nz = (fx[1]-fx[0])*(fy[2]-fy[0]) - (fy[1]-fy[0])*(fx[2]-fx[0]);
    float ix[3], iy[3];
    #pragma unroll
    for (int k = 0; k < 3; ++k) {
        float s = FOCAL_F / (-fz3[k]);
        ix[k] = fx[k]*s; iy[k] = fy[k]*s;
    }
    float ax = ix[0], ay = iy[0], bx = ix[1], by = iy[1], cx = ix[2], cy = iy[2];
    float denom = (bx-ax)*(cy-ay) - (by-ay)*(cx-ax);
    float dsafe = (fabsf(denom) > 1e-9f) ? denom : 1e-9f;
    float inv = 1.0f / dsafe;
    float w0x = (by-cy)*inv, w0y = (cx-bx)*inv, w0c = (bx*cy - by*cx)*inv;
    float w1x = (cy-ay)*inv, w1y = (ax-cx)*inv, w1c = (cx*ay - cy*ax)*inv;
    float w2x = (ay-by)*inv, w2y = (bx-ax)*inv, w2c = (ax*by - ay*bx)*inv;
    float zx = w0x*fz3[0] + w1x*fz3[1] + w2x*fz3[2];
    float zy = w0y*fz3[0] + w1y*fz3[1] + w2y*fz3[2];
    float zc = w0c*fz3[0] + w1c*fz3[1] + w2c*fz3[2];
    bool valid = (nz > 0.0f) && (fabsf(denom) > 1e-9f);
    // fold invalid faces into the w0 plane
    w0x = valid ? w0x : 0.0f;
    w0y = valid ? w0y : 0.0f;
    w0c = valid ? w0c : -1e30f;

    int g = face >> 2, fig = face & 3;
    float4* o = (float4*)(bOp + (size_t)r * Fn * 16) + (size_t)g * 16;
    o[fig*2 + 0]     = make_float4(w0x, w0y, w1x, w1y);
    o[fig*2 + 1]     = make_float4(w2x, w2y, zx,  zy);
    o[8 + fig*2 + 0] = make_float4(w0c, 0.0f, w1c, 0.0f);
    o[8 + fig*2 + 1] = make_float4(w2c, 0.0f, zc,  0.0f);
}

// ------------------------------------------------------------- rasterize ----
// Wave handles a 16-pixel strip. Faces processed 4 at a time via one
// V_WMMA_F32_16X16X4_F32:  D[16 pix x 16 attrs] = A[16x4 (px,py,1,0)] x B[4x16].
// D layout: lane n<16 -> rows 0..7 = faces {4g,4g+1}; lane>=16 -> faces {4g+2,4g+3}.
__global__ __launch_bounds__(256) void k_raster(const float* __restrict__ bOp,
                                                const float* __restrict__ ffeat,
                                                float* __restrict__ featsBuf,
                                                float* __restrict__ softBuf) {
    __shared__ __align__(16) v2f ldsB[(Fn/4) * 32];   // 40 KB, lane-swizzled B ops
    const int r = blockIdx.y;
    {
        const float4* src = (const float4*)(bOp + (size_t)r * Fn * 16);
        float4* dst = (float4*)ldsB;
        for (int i = threadIdx.x; i < Fn * 4; i += 256) dst[i] = src[i];
    }
    __syncthreads();

    const int lane  = threadIdx.x & 31;
    const int wv    = threadIdx.x >> 5;
    const int strip = blockIdx.x * 8 + wv;          // 0..1599
    const int row   = strip / (Ww/16);
    const int col0  = (strip % (Ww/16)) * 16;
    const int m     = lane & 15;

    const float px = -1.0f + 2.0f*(float)(col0 + m) / (float)(Ww - 1);
    const float py =  1.0f - 2.0f*(float)row / (float)(Hh - 1);
    v2f a;
    a.x = (lane < 16) ? px : 1.0f;
    a.y = (lane < 16) ? py : 0.0f;

    float bestZ = -1e9f, bw0 = 0.0f, bw1 = 0.0f, bw2 = 0.0f;
    int   bestIdx = 0;
    float maxArg = -3.0e38f;
    const int halfsel = lane >> 4;

    #pragma unroll 2
    for (int g = 0; g < Fn/4; ++g) {
        v2f b = ldsB[g*32 + lane];                  // one conflict-free ds_load_b64
        v8f c = {0.0f,0.0f,0.0f,0.0f,0.0f,0.0f,0.0f,0.0f};
        v8f d = __builtin_amdgcn_wmma_f32_16x16x4_f32(false, a, false, b,
                                                      (short)0, c, false, false);
        const int fA = g*4 + halfsel*2;
        #pragma unroll
        for (int h = 0; h < 2; ++h) {
            float w0 = d[h*4+0], w1 = d[h*4+1], w2 = d[h*4+2], z = d[h*4+3];
            int   fi = fA + h;
            float mw = fminf(w0, fminf(w1, w2));
            maxArg = fmaxf(maxArg, mw);             // invalid faces folded to -1e30
            int inside = (w0 >= -1e-6f) & (w1 >= -1e-6f) & (w2 >= -1e-6f);
            int take = inside & ((z > bestZ) | ((z == bestZ) & (fi < bestIdx)));
            bestZ   = take ? z  : bestZ;
            bestIdx = take ? fi : bestIdx;
            bw0 = take ? w0 : bw0;
            bw1 = take ? w1 : bw1;
            bw2 = take ? w2 : bw2;
        }
    }
    // merge the two half-waves (same pixel, disjoint face subsets)
    float oz = __shfl_xor(bestZ, 16, 32);
    int   oi = __shfl_xor(bestIdx, 16, 32);
    float o0 = __shfl_xor(bw0, 16, 32);
    float o1 = __shfl_xor(bw1, 16, 32);
    float o2 = __shfl_xor(bw2, 16, 32);
    float oa = __shfl_xor(maxArg, 16, 32);
    int take = (oz > bestZ) | ((oz == bestZ) & (oi < bestIdx));
    bestZ   = take ? oz : bestZ;
    bestIdx = take ? oi : bestIdx;
    bw0 = take ? o0 : bw0;
    bw1 = take ? o1 : bw1;
    bw2 = take ? o2 : bw2;
    maxArg = fmaxf(maxArg, oa);

    if (lane < 16) {
        float hard = (bestZ > -1e8f) ? 1.0f : 0.0f;
        float soft = 1.0f / (1.0f + expf(-SIGMAINV_F * maxArg));
        int pix = row*Ww + col0 + m;
        const float* ff = ffeat + (size_t)bestIdx * 9;
        featsBuf[((size_t)r*3 + 0)*HW + pix] = hard*(bw0*ff[0] + bw1*ff[3] + bw2*ff[6]);
        featsBuf[((size_t)r*3 + 1)*HW + pix] = hard*(bw0*ff[1] + bw1*ff[4] + bw2*ff[7]);
        featsBuf[((size_t)r*3 + 2)*HW + pix] = hard*(bw0*ff[2] + bw1*ff[5] + bw2*ff[8]);
        softBuf[(size_t)r*HW + pix] = soft;
    }
}

// --------------------------------------------------------------- erode -----
__global__ void k_erode(const float* __restrict__ src, float* __restrict__ dst) {
    int idx = blockIdx.x * blockDim.x + threadIdx.x;
    if (idx >= NR * HW) return;
    int r = idx / HW, p = idx % HW, h = p / Ww, w = p % Ww;
    const float* s = src + (size_t)r*HW;
    float mn = 1e30f;
    for (int dh = -1; dh <= 1; ++dh)
        for (int dw = -1; dw <= 1; ++dw) {
            int hh = h + dh, ww = w + dw;
            if (hh >= 0 && hh < Hh && ww >= 0 && ww < Ww)
                mn = fminf(mn, s[hh*Ww + ww]);
        }
    dst[idx] = mn;
}

// --------------------------------------------------------------- blurs -----
__global__ void k_blur_v(const float* __restrict__ src, float* __restrict__ dst, int total) {
    int idx = blockIdx.x * blockDim.x + threadIdx.x;
    if (idx >= total) return;
    int img = idx / HW, p = idx % HW, h = p / Ww, w = p % Ww;
    const float* s = src + (size_t)img*HW;
    float sum = 0.0f;
    #pragma unroll
    for (int k = 0; k < 11; ++k) {
        int hh = h + k - 5;
        if (hh < 0) hh = -hh;
        if (hh >= Hh) hh = 2*Hh - 2 - hh;
        sum += G11[k] * s[hh*Ww + w];
    }
    dst[idx] = sum;
}

__device__ inline float blur_row(const float* s, int h, int w) {
    float sum = 0.0f;
    #pragma unroll
    for (int k = 0; k < 11; ++k) {
        int ww = w + k - 5;
        if (ww < 0) ww = -ww;
        if (ww >= Ww) ww = 2*Ww - 2 - ww;
        sum += G11[k] * s[h*Ww + ww];
    }
    return sum;
}

__global__ void k_blur_h(const float* __restrict__ src, float* __restrict__ dst, int total) {
    int idx = blockIdx.x * blockDim.x + threadIdx.x;
    if (idx >= total) return;
    int img = idx / HW, p = idx % HW;
    dst[idx] = blur_row(src + (size_t)img*HW, p / Ww, p % Ww);
}

__global__ void k_blur_h_rgb_out(const float* __restrict__ src, float* __restrict__ out) {
    int idx = blockIdx.x * blockDim.x + threadIdx.x;
    if (idx >= NR*3*HW) return;
    int img = idx / HW, p = idx % HW;
    int r = img / 3, c = img % 3;
    out[((size_t)(r*4 + c))*HW + p] = blur_row(src + (size_t)img*HW, p / Ww, p % Ww);
}

__global__ void k_blur_h_mask_out(const float* __restrict__ src, float* __restrict__ out) {
    int idx = blockIdx.x * blockDim.x + threadIdx.x;
    if (idx >= NR*HW) return;
    int r = idx / HW, p = idx % HW;
    out[((size_t)(r*4 + 3))*HW + p] = blur_row(src + (size_t)r*HW, p / Ww, p % Ww);
}

// --------------------------------------------------------------- launch ----
extern "C" void kernel_launch(void* const* d_in, const int* in_sizes, int n_in,
                              void* d_out, int out_size, void* d_ws, size_t ws_size,
                              hipStream_t stream) {
    const float* trans  = (const float*)d_in[0];   // (1,1,2,2,3)
    const float* quat   = (const float*)d_in[1];   // (1,2,2,4)
    const float* uverts = (const float*)d_in[2];   // (1,642,3)
    const float* ffeat  = (const float*)d_in[3];   // (1,640,3,3)
    const int*   faces  = (const int*)d_in[4];     // (640,3)
    float* out = (float*)d_out;
    float* ws  = (float*)d_ws;

    float* Rbuf      = ws;                         // 256
    float* bOp       = ws + 256;                   // 16*640*16 = 163840
    float* featsBuf  = ws + 164096;                // 48*HW    = 1228800
    float* softBuf   = ws + 1392896;               // 16*HW
    float* maskE     = ws + 1802496;               // 16*HW
    float* mtmp1     = ws + 2212096;               // 16*HW
    float* mtmp2     = ws + 2621696;               // 16*HW
    float* tmpRGB    = ws + 3031296;               // 48*HW

    k_setup<<<1, 32, 0, stream>>>(trans, quat, Rbuf);
    k_facesetup<<<(NR*Fn + 255)/256, 256, 0, stream>>>(Rbuf, uverts, faces, bOp);
    k_raster<<<dim3((HW/16)/8, NR), 256, 0, stream>>>(bOp, ffeat, featsBuf, softBuf);
    k_erode<<<(NR*HW + 255)/256, 256, 0, stream>>>(softBuf, maskE);
    // rgb: blur once
    k_blur_v<<<(NR*3*HW + 255)/256, 256, 0, stream>>>(featsBuf, tmpRGB, NR*3*HW);
    k_blur_h_rgb_out<<<(NR*3*HW + 255)/256, 256, 0, stream>>>(tmpRGB, out);
    // mask: blur twice
    k_blur_v<<<(NR*HW + 255)/256, 256, 0, stream>>>(maskE, mtmp1, NR*HW);
    k_blur_h<<<(NR*HW + 255)/256, 256, 0, stream>>>(mtmp1, mtmp2, NR*HW);
    k_blur_v<<<(NR*HW + 255)/256, 256, 0, stream>>>(mtmp2, mtmp1, NR*HW);
    k_blur_h_mask_out<<<(NR*HW + 255)/256, 256, 0, stream>>>(mtmp1, out);
}